// PointNetSetAbstraction_50568944943630
// MI455X (gfx1250) — compile-verified
//
#include <hip/hip_runtime.h>
#include <hip/hip_bf16.h>

typedef float v2f __attribute__((ext_vector_type(2)));
typedef float v8f __attribute__((ext_vector_type(8)));

#define NPT   16384
#define NSEG  512     // npoint S
#define KS    32      // nsample K
#define NB    8       // batch
#define MCOL  131072  // B*K*S columns of the MLP GEMMs

// ---------------------------------------------------------------------------
// 1) Farthest point sampling: one block per batch, 1024 threads, 16 pts/thread
//    in registers; LDS argmax reduction with first-index tie-break.
//    Writes new_xyz (B,3,512) directly into d_out.
// ---------------------------------------------------------------------------
#define FPS_T 1024
__global__ void fps_kernel(const float* __restrict__ xyz, float* __restrict__ out_newxyz) {
    const int N = NPT, S = NSEG;
    int b = blockIdx.x;
    int t = threadIdx.x;
    const float* xb = xyz + (size_t)b * 3 * N;

    float px[16], py[16], pz[16], dist[16];
#pragma unroll
    for (int j = 0; j < 16; ++j) {
        int n = j * FPS_T + t;
        px[j] = xb[n];
        py[j] = xb[N + n];
        pz[j] = xb[2 * N + n];
        dist[j] = 1e10f;
    }

    __shared__ float bd[FPS_T];
    __shared__ int   bi[FPS_T];

    int farthest = 0;
    for (int s = 0; s < S; ++s) {
        float cx = xb[farthest];
        float cy = xb[N + farthest];
        float cz = xb[2 * N + farthest];
        if (t == 0) {
            out_newxyz[(size_t)b * 3 * S + 0 * S + s] = cx;
            out_newxyz[(size_t)b * 3 * S + 1 * S + s] = cy;
            out_newxyz[(size_t)b * 3 * S + 2 * S + s] = cz;
        }
        float bestd = -1.0f;
        int   besti = 0x7fffffff;
#pragma unroll
        for (int j = 0; j < 16; ++j) {
            float dx = px[j] - cx, dy = py[j] - cy, dz = pz[j] - cz;
            float d = dx * dx + dy * dy + dz * dz;
            float nd = fminf(dist[j], d);
            dist[j] = nd;
            int n = j * FPS_T + t;
            if (nd > bestd) { bestd = nd; besti = n; }
        }
        bd[t] = bestd;
        bi[t] = besti;
        __syncthreads();
        for (int off = FPS_T / 2; off > 0; off >>= 1) {
            if (t < off) {
                float od = bd[t + off];
                int   oi = bi[t + off];
                if (od > bd[t] || (od == bd[t] && oi < bi[t])) { bd[t] = od; bi[t] = oi; }
            }
            __syncthreads();
        }
        farthest = bi[0];
        __syncthreads();
    }
}

// ---------------------------------------------------------------------------
// 2) Ball query + grouping: one wave32 block per (b,s). Ballot + prefix
//    popcount appends the first K in-radius indices in ascending order,
//    matching reference sort semantics. Fuses gather + centering, writing
//    x0 in [C=8][MCOL] layout (rows 6,7 zero-padded so layer-0 GEMM can run
//    an unguarded K=8 loop) with column m = b*16384 + k*512 + s.
// ---------------------------------------------------------------------------
__global__ void ballgroup_kernel(const float* __restrict__ xyz,
                                 const float* __restrict__ pts,
                                 const float* __restrict__ newxyz,
                                 float* __restrict__ X0) {
    const int N = NPT, S = NSEG;
    int q = blockIdx.x;        // 0..4095
    int b = q >> 9;
    int s = q & 511;
    int lane = threadIdx.x;    // blockDim = 32 (one wave)

    const float* xb = xyz + (size_t)b * 3 * N;
    const float* pb = pts + (size_t)b * 3 * N;
    float cx = newxyz[(size_t)b * 3 * S + 0 * S + s];
    float cy = newxyz[(size_t)b * 3 * S + 1 * S + s];
    float cz = newxyz[(size_t)b * 3 * S + 2 * S + s];
    const float r2 = 0.2f * 0.2f;

    __shared__ int idx_sh[KS];
    int cnt = 0;
    for (int base = 0; base < N && cnt < KS; base += 32) {
        int n = base + lane;
        float dx = xb[n] - cx, dy = xb[N + n] - cy, dz = xb[2 * N + n] - cz;
        float d2 = dx * dx + dy * dy + dz * dz;
        bool pred = d2 <= r2;
        unsigned mask = (unsigned)__builtin_amdgcn_ballot_w32(pred);
        int pos = cnt + __popc(mask & ((1u << lane) - 1u));
        if (pred && pos < KS) idx_sh[pos] = n;
        cnt += __popc(mask);
    }
    __syncthreads();

    int actual = cnt < KS ? cnt : KS;   // >=1 always: the centroid itself is in-radius
    int first = idx_sh[0];
    int k = lane;
    int gi = (k < actual) ? idx_sh[k] : first;

    size_t m = (size_t)b * ((size_t)KS * S) + (size_t)k * S + s;
    X0[0 * (size_t)MCOL + m] = pb[gi];
    X0[1 * (size_t)MCOL + m] = pb[N + gi];
    X0[2 * (size_t)MCOL + m] = pb[2 * N + gi];
    X0[3 * (size_t)MCOL + m] = xb[gi] - cx;
    X0[4 * (size_t)MCOL + m] = xb[N + gi] - cy;
    X0[5 * (size_t)MCOL + m] = xb[2 * N + gi] - cz;
    X0[6 * (size_t)MCOL + m] = 0.0f;    // zero pad rows 6,7
    X0[7 * (size_t)MCOL + m] = 0.0f;
}

// ---------------------------------------------------------------------------
// 2b) Zero-pad layer-0 weights (64x6) into (64x8) so the GEMM needs no guards.
// ---------------------------------------------------------------------------
__global__ void padw0_kernel(const float* __restrict__ w0, float* __restrict__ w0p) {
    int i = blockIdx.x * 64 + threadIdx.x;   // 512 entries
    int o = i >> 3;
    int c = i & 7;
    w0p[i] = (c < 6) ? w0[o * 6 + c] : 0.0f;
}

// ---------------------------------------------------------------------------
// 3) GEMM via V_WMMA_F32_16X16X4_F32. Y[o][m] = sum_c W[o][c]*X[c][m] + bias.
//    CIN is a compile-time template parameter (multiple of 4): the K loop is
//    fully unrolled with no guards -> straight-line loads + wmma stream.
//    One 16x16 D tile per wave; 8 waves (m-tiles) per block.
//    A (16x4 f32): lane l<16 -> M=l, {K0,K1}; lane 16+l -> M=l, {K2,K3}
//    B (4x16 f32): lane n<16 -> N=n, {K0,K1}; lane 16+n -> N=n, {K2,K3}
//    D: VGPR v, lanes 0-15 -> M=v, lanes 16-31 -> M=v+8, N=lane&15
// ---------------------------------------------------------------------------
template <int CIN>
__global__ void gemm_wmma(const float* __restrict__ X, const float* __restrict__ W,
                          const float* __restrict__ bias, float* __restrict__ Y) {
    const int M = MCOL;
    int lane = threadIdx.x & 31;
    int wave = threadIdx.x >> 5;
    int m0 = (blockIdx.x * 8 + wave) * 16;
    int o0 = blockIdx.y * 16;
    int half = lane >> 4;   // 0 or 1
    int l = lane & 15;
    int kbase = half * 2;   // 0 or 2

    const float* wrow = W + (size_t)(o0 + l) * CIN + kbase;   // {K0,K1} / {K2,K3}
    const float* xcol = X + (size_t)kbase * M + m0 + l;

    v8f acc = {};
#pragma unroll
    for (int c0 = 0; c0 < CIN; c0 += 4) {
        v2f a, bm;
        a.x  = wrow[c0];
        a.y  = wrow[c0 + 1];
        bm.x = xcol[(size_t)c0 * M];
        bm.y = xcol[(size_t)(c0 + 1) * M];
        acc = __builtin_amdgcn_wmma_f32_16x16x4_f32(
            /*neg_a=*/false, a, /*neg_b=*/false, bm,
            /*c_mod=*/(short)0, acc, /*reuse_a=*/false, /*reuse_b=*/false);
    }
#pragma unroll
    for (int v = 0; v < 8; ++v) {
        int row = v + half * 8;
        Y[(size_t)(o0 + row) * M + m0 + l] = acc[v] + bias[o0 + row];
    }
}

// ---------------------------------------------------------------------------
// 4) Batchnorm stats: one block per channel, deterministic fixed-tree reduce.
// ---------------------------------------------------------------------------
__global__ void stats_kernel(const float* __restrict__ Y,
                             float* __restrict__ meanv, float* __restrict__ rstdv) {
    const int M = MCOL;
    int o = blockIdx.x;
    int t = threadIdx.x;   // 256
    const float* y = Y + (size_t)o * M;
    float s = 0.0f, s2 = 0.0f;
    for (int i = t; i < M; i += 256) {
        float v = y[i];
        s += v;
        s2 += v * v;
    }
    __shared__ float ss[256], ss2[256];
    ss[t] = s; ss2[t] = s2;
    __syncthreads();
    for (int off = 128; off > 0; off >>= 1) {
        if (t < off) { ss[t] += ss[t + off]; ss2[t] += ss2[t + off]; }
        __syncthreads();
    }
    if (t == 0) {
        float mean = ss[0] / (float)M;
        float var = ss2[0] / (float)M - mean * mean;
        if (var < 0.0f) var = 0.0f;
        meanv[o] = mean;
        rstdv[o] = rsqrtf(var + 1e-5f);
    }
}

// ---------------------------------------------------------------------------
// 5) Normalize + scale/shift + ReLU, in place.
// ---------------------------------------------------------------------------
__global__ void normrelu_kernel(float* __restrict__ Y,
                                const float* __restrict__ meanv,
                                const float* __restrict__ rstdv,
                                const float* __restrict__ g,
                                const float* __restrict__ be) {
    int i = blockIdx.x * 256 + threadIdx.x;
    int o = i >> 17;   // / MCOL
    float v = (Y[i] - meanv[o]) * rstdv[o] * g[o] + be[o];
    Y[i] = v > 0.0f ? v : 0.0f;
}

// ---------------------------------------------------------------------------
// 6) Max over K -> new_feat (B,128,512), appended after new_xyz in d_out.
// ---------------------------------------------------------------------------
__global__ void maxk_kernel(const float* __restrict__ Y, float* __restrict__ out) {
    const int M = MCOL, S = NSEG, O = 128;
    int i = blockIdx.x * 256 + threadIdx.x;   // B*O*S = 524288
    int s = i % S;
    int o = (i / S) % O;
    int b = i / (S * O);
    const float* y = Y + (size_t)o * M + (size_t)b * ((size_t)KS * S) + s;
    float mv = -1e30f;
#pragma unroll
    for (int k = 0; k < KS; ++k) mv = fmaxf(mv, y[(size_t)k * S]);
    out[i] = mv;
}

// ---------------------------------------------------------------------------
extern "C" void kernel_launch(void* const* d_in, const int* in_sizes, int n_in,
                              void* d_out, int out_size, void* d_ws, size_t ws_size,
                              hipStream_t stream) {
    const float* xyz = (const float*)d_in[0];
    const float* pts = (const float*)d_in[1];
    const float* w0  = (const float*)d_in[2];
    const float* b0  = (const float*)d_in[3];
    const float* g0  = (const float*)d_in[4];
    const float* be0 = (const float*)d_in[5];
    const float* w1  = (const float*)d_in[6];
    const float* b1  = (const float*)d_in[7];
    const float* g1  = (const float*)d_in[8];
    const float* be1 = (const float*)d_in[9];
    const float* w2  = (const float*)d_in[10];
    const float* b2  = (const float*)d_in[11];
    const float* g2  = (const float*)d_in[12];
    const float* be2 = (const float*)d_in[13];

    float* out = (float*)d_out;
    float* out_newxyz = out;                     // (8,3,512)
    float* out_feat   = out + NB * 3 * NSEG;     // (8,128,512)

    // Workspace layout (floats)
    float* X0 = (float*)d_ws;                    //   8 * MCOL (rows 6,7 zero)
    float* Y0 = X0 + (size_t)8  * MCOL;          //  64 * MCOL
    float* Y1 = Y0 + (size_t)64 * MCOL;          //  64 * MCOL
    float* Y2 = Y1 + (size_t)64 * MCOL;          // 128 * MCOL
    float* st = Y2 + (size_t)128 * MCOL;         // stats: 6 * 128
    float* mean0 = st;        float* rstd0 = st + 128;
    float* mean1 = st + 256;  float* rstd1 = st + 384;
    float* mean2 = st + 512;  float* rstd2 = st + 640;
    float* w0p   = st + 768;                     // 64 x 8 padded layer-0 weights

    // 1) FPS -> new_xyz (in d_out)
    fps_kernel<<<NB, FPS_T, 0, stream>>>(xyz, out_newxyz);

    // 2) Pad layer-0 weights; ball query + group -> X0 [8][MCOL]
    padw0_kernel<<<8, 64, 0, stream>>>(w0, w0p);
    ballgroup_kernel<<<NB * NSEG, 32, 0, stream>>>(xyz, pts, out_newxyz, X0);

    // 3) Layer 0: 6(->8) -> 64
    gemm_wmma<8><<<dim3(MCOL / 128, 64 / 16), 256, 0, stream>>>(X0, w0p, b0, Y0);
    stats_kernel<<<64, 256, 0, stream>>>(Y0, mean0, rstd0);
    normrelu_kernel<<<(64 * MCOL) / 256, 256, 0, stream>>>(Y0, mean0, rstd0, g0, be0);

    // 4) Layer 1: 64 -> 64
    gemm_wmma<64><<<dim3(MCOL / 128, 64 / 16), 256, 0, stream>>>(Y0, w1, b1, Y1);
    stats_kernel<<<64, 256, 0, stream>>>(Y1, mean1, rstd1);
    normrelu_kernel<<<(64 * MCOL) / 256, 256, 0, stream>>>(Y1, mean1, rstd1, g1, be1);

    // 5) Layer 2: 64 -> 128
    gemm_wmma<64><<<dim3(MCOL / 128, 128 / 16), 256, 0, stream>>>(Y1, w2, b2, Y2);
    stats_kernel<<<128, 256, 0, stream>>>(Y2, mean2, rstd2);
    normrelu_kernel<<<(128 * MCOL) / 256, 256, 0, stream>>>(Y2, mean2, rstd2, g2, be2);

    // 6) Max over K -> new_feat
    maxk_kernel<<<(NB * 128 * NSEG) / 256, 256, 0, stream>>>(Y2, out_feat);
}